// TopKGate_3874060501188
// MI455X (gfx1250) — compile-verified
//
#include <hip/hip_runtime.h>
#include <hip/hip_bf16.h>
#include <cstdint>
#include <cstddef>

// ---------------------------------------------------------------------------
// TopKGate (top-1 MoE gating) for MI455X / gfx1250.
// Store-bandwidth bound: 537MB of output zeros (~23us @ 23.3TB/s) dominates;
// gate GEMM (134MB x-read, 4.3 GFLOP) uses fp32 WMMA (V_WMMA_F32_16X16X4_F32)
// to match the reference's forced-fp32 gate projection.
// ---------------------------------------------------------------------------

typedef float v2f __attribute__((ext_vector_type(2)));
typedef float v4f __attribute__((ext_vector_type(4)));
typedef float v8f __attribute__((ext_vector_type(8)));

#define S_TOKENS   8192
#define N_EXPERTS  64
#define DMODEL     4096
#define CAPACITY   128
#define COMBINE_ELEMS 67108864ull   // 8192*64*128

// ---------------- kernel 1: gate GEMM via fp32 WMMA ------------------------
// grid = 512 (16 token-rows per block), block = 128 (4 waves).
// Wave w computes tile rows [16*bx, 16*bx+16) x experts [16w, 16w+16).
// A fragment (16x4 f32): lane L<16 holds row M=L, K={0,1}; lane L+16 holds K={2,3}.
// B fragment (4x16 f32): lane L<16 holds col N=L, K={0,1}; lane L+16 holds K={2,3}.
// B[k][n] = wg[n][k] (wg is [64,4096] row-major), so B loads stream wg rows.
__global__ __launch_bounds__(128)
void gate_gemm_kernel(const float* __restrict__ x,
                      const float* __restrict__ wg,
                      float* __restrict__ logits) {
    const int wave = threadIdx.x >> 5;
    const int lane = threadIdx.x & 31;
    const int r0   = blockIdx.x * 16;
    const int e0   = wave * 16;
    const int half = lane >> 4;      // 0: K pair {0,1}; 1: K pair {2,3}
    const int l15  = lane & 15;

    const float* xp = x  + (size_t)(r0 + l15) * DMODEL + 2 * half;
    const float* wp = wg + (size_t)(e0 + l15) * DMODEL + 2 * half;

    __builtin_prefetch(xp, 0, 0);
    __builtin_prefetch(wp, 0, 0);

    v8f acc = {};
#pragma unroll 8
    for (int k = 0; k < DMODEL; k += 4) {
        v2f a = *(const v2f*)(xp + k);   // global_load_b64
        v2f b = *(const v2f*)(wp + k);
        acc = __builtin_amdgcn_wmma_f32_16x16x4_f32(
            /*neg_a=*/false, a, /*neg_b=*/false, b,
            /*c_mod=*/(short)0, acc, /*reuse_a=*/false, /*reuse_b=*/false);
    }
    // C/D layout: lane L -> N = L&15, M = (L>>4)*8 + i for VGPR i.
    const int n  = l15;
    const int mb = half * 8;
#pragma unroll
    for (int i = 0; i < 8; ++i)
        logits[(size_t)(r0 + mb + i) * N_EXPERTS + e0 + n] = acc[i];
}

// ---------------- kernel 0: zero tiny accumulators -------------------------
__global__ void init_ws_kernel(int* __restrict__ counts, float* __restrict__ me_acc) {
    const int t = threadIdx.x;
    if (t < N_EXPERTS) { counts[t] = 0; me_acc[t] = 0.0f; }
}

// ---------------- kernel 2: softmax + argmax per token ---------------------
// One wave per token (wave32: each lane owns experts {lane, lane+32}).
__global__ __launch_bounds__(256)
void softmax_argmax_kernel(const float* __restrict__ logits,
                           const float* __restrict__ noise,
                           int*   __restrict__ idx1,
                           float* __restrict__ gate1,
                           float* __restrict__ noise1,
                           int*   __restrict__ counts,
                           float* __restrict__ me_acc) {
    __shared__ float sm[N_EXPERTS];
    const int tid = threadIdx.x;
    if (tid < N_EXPERTS) sm[tid] = 0.0f;
    __syncthreads();

    const int wave = tid >> 5, lane = tid & 31;
    const int s = blockIdx.x * 8 + wave;

    const float l0 = logits[(size_t)s * N_EXPERTS + lane];
    const float l1 = logits[(size_t)s * N_EXPERTS + lane + 32];

    float m = fmaxf(l0, l1);
    for (int off = 16; off > 0; off >>= 1) m = fmaxf(m, __shfl_xor(m, off));
    const float p0 = __expf(l0 - m), p1 = __expf(l1 - m);
    float sum = p0 + p1;
    for (int off = 16; off > 0; off >>= 1) sum += __shfl_xor(sum, off);
    const float inv = 1.0f / sum;
    const float g0 = p0 * inv, g1 = p1 * inv;

    // argmax, first-index tie break (jnp.argmax semantics)
    float bv; int bi;
    if (l0 >= l1) { bv = l0; bi = lane; } else { bv = l1; bi = lane + 32; }
    for (int off = 16; off > 0; off >>= 1) {
        const float ov = __shfl_xor(bv, off);
        const int   oi = __shfl_xor(bi, off);
        if (ov > bv || (ov == bv && oi < bi)) { bv = ov; bi = oi; }
    }
    if (lane == 0) {
        idx1[s]   = bi;
        gate1[s]  = __expf(bv - m) * inv;                  // gates[s, argmax]
        noise1[s] = noise[(size_t)s * N_EXPERTS + bi];     // mask1_rand nonzero entry
        atomicAdd(&counts[bi], 1);
    }
    atomicAdd(&sm[lane], g0);          // ds_add_f32 (LDS float atomics)
    atomicAdd(&sm[lane + 32], g1);
    __syncthreads();
    if (tid < N_EXPERTS) atomicAdd(&me_acc[tid], sm[tid]);
}

// ---------------- kernel 3: per-expert capacity selection + slot cumsum ----
// One block per expert. Top-128-by-noise via binary search on the float bit
// pattern (monotone for noise in [0,1)), then an ordered ballot-scan to
// assign cumsum slots in token-index order (reference's cumsum semantics).
__global__ __launch_bounds__(256)
void capacity_kernel(const int* __restrict__ idx1,
                     const float* __restrict__ noise1,
                     const int* __restrict__ counts,
                     int* __restrict__ slot1) {
    const int e   = blockIdx.x;
    const int tid = threadIdx.x;
    __shared__ int cnt;
    __shared__ int wcnt[8];
    __shared__ int base_s;

    const int total = counts[e];
    unsigned thresh = 0u;                    // total<=cap: select every assigned token
    if (total > CAPACITY) {
        // smallest u with count(bits >= u) <= CAPACITY  ->  selects top-128 exactly
        unsigned lo = 0u, hi = 0x3F800000u;  // bits(1.0f); count(>=1.0) == 0
        while (lo < hi) {
            const unsigned mid = lo + ((hi - lo) >> 1);
            if (tid == 0) cnt = 0;
            __syncthreads();
            int local = 0;
            for (int s = tid; s < S_TOKENS; s += 256)
                if (idx1[s] == e && __float_as_uint(noise1[s]) >= mid) ++local;
            atomicAdd(&cnt, local);
            __syncthreads();
            const int f = cnt;
            __syncthreads();
            if (f <= CAPACITY) hi = mid; else lo = mid + 1;
        }
        thresh = hi;
    }

    if (tid == 0) base_s = 0;
    __syncthreads();
    const int wave = tid >> 5, lane = tid & 31;
    const unsigned lmask = (1u << lane) - 1u;
    for (int c0 = 0; c0 < S_TOKENS; c0 += 256) {
        const int  s    = c0 + tid;
        const bool mine = (idx1[s] == e);
        const bool flag = mine && (__float_as_uint(noise1[s]) >= thresh);
        const unsigned bm = (unsigned)__ballot(flag);      // wave32 mask
        if (lane == 0) wcnt[wave] = __popc(bm);
        __syncthreads();
        int pre = base_s;
        for (int i = 0; i < wave; ++i) pre += wcnt[i];
        const int slot = pre + __popc(bm & lmask);
        if (mine) slot1[s] = (flag && slot < CAPACITY) ? slot : -1; // unique writer per token
        __syncthreads();
        if (tid == 0) {
            int t = base_s;
            for (int i = 0; i < 8; ++i) t += wcnt[i];
            base_s = t;
        }
        __syncthreads();
    }
}

// ---------------- kernel 4: streaming zero-fill of d_out -------------------
// 537MB of zeros -> the dominant cost (~23us @ 23.3 TB/s). Nontemporal b128
// stores via a native clang ext-vector (HIP's float4 struct is rejected by
// __builtin_nontemporal_store).
__global__ __launch_bounds__(256)
void zero_out_kernel(float* __restrict__ out, size_t n) {
    const size_t stride = (size_t)gridDim.x * blockDim.x;
    const size_t i = (size_t)blockIdx.x * blockDim.x + threadIdx.x;
    const size_t n4 = n >> 2;
    v4f* p4 = (v4f*)out;
    const v4f z = {0.f, 0.f, 0.f, 0.f};
    for (size_t j = i; j < n4; j += stride)
        __builtin_nontemporal_store(z, p4 + j);
    if (i == 0)
        for (size_t j = n4 << 2; j < n; ++j) out[j] = 0.0f;
}

// ---------------- kernel 5: sparse scatter of nonzeros ---------------------
__global__ __launch_bounds__(256)
void scatter_kernel(const int* __restrict__ idx1, const float* __restrict__ gate1,
                    const int* __restrict__ slot1, float* __restrict__ out) {
    const int s = blockIdx.x * blockDim.x + threadIdx.x;
    if (s >= S_TOKENS) return;
    const int sl = slot1[s];
    if (sl >= 0) {
        const int e = idx1[s];
        const size_t off = 1ull + (((size_t)s * N_EXPERTS + e) << 7) + (size_t)sl;
        out[off]                 = gate1[s];   // combine_weights[s,e,c]
        out[off + COMBINE_ELEMS] = 1.0f;       // dispatch_mask[s,e,c]
    }
}

// ---------------- kernel 6: l_aux + exp_counts -----------------------------
__global__ void finalize_kernel(const int* __restrict__ counts,
                                const float* __restrict__ me_acc,
                                float* __restrict__ out) {
    __shared__ float terms[N_EXPERTS];
    const int e = threadIdx.x;
    const float me = me_acc[e] * (1.0f / S_TOKENS);
    const float ce = (float)counts[e] * (1.0f / S_TOKENS);
    terms[e] = me * ce;
    out[1ull + 2ull * COMBINE_ELEMS + e] = (float)counts[e];  // exp_counts
    __syncthreads();
    if (e == 0) {
        float sum = 0.0f;
        for (int i = 0; i < N_EXPERTS; ++i) sum += terms[i];
        out[0] = sum * (float)N_EXPERTS;                      // l_aux
    }
}

// ---------------------------------------------------------------------------
extern "C" void kernel_launch(void* const* d_in, const int* in_sizes, int n_in,
                              void* d_out, int out_size, void* d_ws, size_t ws_size,
                              hipStream_t stream) {
    const float* x     = (const float*)d_in[0];   // [8192, 4096]
    const float* wg    = (const float*)d_in[1];   // [64, 4096]
    const float* noise = (const float*)d_in[2];   // [8192, 64]
    float* out = (float*)d_out;

    char* ws = (char*)d_ws;
    float* logits = (float*)ws;  ws += (size_t)S_TOKENS * N_EXPERTS * sizeof(float);
    int*   idx1   = (int*)ws;    ws += (size_t)S_TOKENS * sizeof(int);
    float* gate1  = (float*)ws;  ws += (size_t)S_TOKENS * sizeof(float);
    float* noise1 = (float*)ws;  ws += (size_t)S_TOKENS * sizeof(float);
    int*   slot1  = (int*)ws;    ws += (size_t)S_TOKENS * sizeof(int);
    int*   counts = (int*)ws;    ws += (size_t)N_EXPERTS * sizeof(int);
    float* me_acc = (float*)ws;

    const size_t total_out = 1ull + 2ull * COMBINE_ELEMS + (size_t)N_EXPERTS;

    init_ws_kernel      <<<1, 64, 0, stream>>>(counts, me_acc);
    gate_gemm_kernel    <<<S_TOKENS / 16, 128, 0, stream>>>(x, wg, logits);
    softmax_argmax_kernel<<<S_TOKENS / 8, 256, 0, stream>>>(logits, noise, idx1, gate1,
                                                            noise1, counts, me_acc);
    capacity_kernel     <<<N_EXPERTS, 256, 0, stream>>>(idx1, noise1, counts, slot1);
    zero_out_kernel     <<<4096, 256, 0, stream>>>(out, total_out);
    scatter_kernel      <<<S_TOKENS / 256, 256, 0, stream>>>(idx1, gate1, slot1, out);
    finalize_kernel     <<<1, N_EXPERTS, 0, stream>>>(counts, me_acc, out);
}